// HomographyWarper_14491219657116
// MI455X (gfx1250) — compile-verified
//
#include <hip/hip_runtime.h>

// HomographyWarper for fixed reference shapes: x (4,64,512,512) f32, H (4,3,3) f32.
// Memory-bound gather: ~512 MB DRAM traffic -> ~22 us roofline @ 23.3 TB/s.
// No matmul structure -> WMMA intentionally unused; kernel minimizes VMEM issue
// (4 loads + 1 NT store per channel) and uses NT stores + global_prefetch_b8
// as the gfx1250 data-movement knobs.

#define Bn  4
#define Cn  64
#define Hn  512
#define Wn  512
#define HWn (Hn * Wn)

__global__ __launch_bounds__(256)
void homography_warp_kernel(const float* __restrict__ x,
                            const float* __restrict__ Hmat,
                            float* __restrict__ out)
{
    const int b = blockIdx.y;
    const int p = blockIdx.x * 256 + threadIdx.x;   // flat pixel id in [0, H*W)
    const int py = p >> 9;                          // p / 512
    const int px = p & (Wn - 1);                    // p % 512

    // ---- load H[b] (row-major 3x3) and invert via adjugate / det ----
    const float* Hb = Hmat + b * 9;
    const float h00 = Hb[0], h01 = Hb[1], h02 = Hb[2];
    const float h10 = Hb[3], h11 = Hb[4], h12 = Hb[5];
    const float h20 = Hb[6], h21 = Hb[7], h22 = Hb[8];

    const float c00 =  (h11 * h22 - h12 * h21);
    const float c01 = -(h10 * h22 - h12 * h20);
    const float c02 =  (h10 * h21 - h11 * h20);
    const float invdet = 1.0f / (h00 * c00 + h01 * c01 + h02 * c02);

    const float i00 =  c00 * invdet;
    const float i01 = -(h01 * h22 - h02 * h21) * invdet;
    const float i02 =  (h01 * h12 - h02 * h11) * invdet;
    const float i10 =  c01 * invdet;
    const float i11 =  (h00 * h22 - h02 * h20) * invdet;
    const float i12 = -(h00 * h12 - h02 * h10) * invdet;
    const float i20 =  c02 * invdet;
    const float i21 = -(h00 * h21 - h01 * h20) * invdet;
    const float i22 =  (h00 * h11 - h01 * h10) * invdet;

    // ---- normalized grid coords (align_corners=True linspace(-1,1,N)) ----
    const float gx = fmaf((float)px, 2.0f / 511.0f, -1.0f);
    const float gy = fmaf((float)py, 2.0f / 511.0f, -1.0f);

    // mapped = H_inv * [gx, gy, 1]
    const float mx = fmaf(i00, gx, fmaf(i01, gy, i02));
    const float my = fmaf(i10, gx, fmaf(i11, gy, i12));
    const float mw = fmaf(i20, gx, fmaf(i21, gy, i22));
    const float rw = 1.0f / (mw + 1e-8f);           // reference EPS
    const float xn = mx * rw;
    const float yn = my * rw;

    // pixel-space sample coords
    const float ix = (xn + 1.0f) * 0.5f * 511.0f;
    const float iy = (yn + 1.0f) * 0.5f * 511.0f;

    const float x0f = floorf(ix), y0f = floorf(iy);
    const float x1f = x0f + 1.0f, y1f = y0f + 1.0f;
    const float wx1 = ix - x0f,  wy1 = iy - y0f;
    const float wx0 = 1.0f - wx1, wy0 = 1.0f - wy1;

    // zeros padding: fold per-tap validity into the bilinear weights
    const float vx0 = (x0f >= 0.0f && x0f <= 511.0f) ? 1.0f : 0.0f;
    const float vx1 = (x1f >= 0.0f && x1f <= 511.0f) ? 1.0f : 0.0f;
    const float vy0 = (y0f >= 0.0f && y0f <= 511.0f) ? 1.0f : 0.0f;
    const float vy1 = (y1f >= 0.0f && y1f <= 511.0f) ? 1.0f : 0.0f;

    const float w00 = wx0 * wy0 * (vx0 * vy0);
    const float w10 = wx1 * wy0 * (vx1 * vy0);
    const float w01 = wx0 * wy1 * (vx0 * vy1);
    const float w11 = wx1 * wy1 * (vx1 * vy1);

    // clamped integer tap coords -> always-in-bounds loads
    const int x0i = (int)fminf(fmaxf(x0f, 0.0f), 511.0f);
    const int x1i = (int)fminf(fmaxf(x1f, 0.0f), 511.0f);
    const int y0i = (int)fminf(fmaxf(y0f, 0.0f), 511.0f);
    const int y1i = (int)fminf(fmaxf(y1f, 0.0f), 511.0f);

    const int o00 = y0i * Wn + x0i;
    const int o10 = y0i * Wn + x1i;
    const int o01 = y1i * Wn + x0i;
    const int o11 = y1i * Wn + x1i;

    const float* __restrict__ xb = x   + (size_t)b * Cn * HWn;
    float*       __restrict__ ob = out + (size_t)b * Cn * HWn + p;

    // warm L2 for a mid-tensor channel plane while early channels execute
    // (lowers to global_prefetch_b8 on gfx1250)
    __builtin_prefetch(xb + 32 * HWn + o00, 0, 1);

    // ---- channel loop: 4 coalesced gathers + 1 NT store per channel ----
#pragma unroll 8
    for (int c = 0; c < Cn; ++c) {
        const float* __restrict__ pl = xb + c * HWn;
        float v = pl[o00] * w00;
        v = fmaf(pl[o10], w10, v);
        v = fmaf(pl[o01], w01, v);
        v = fmaf(pl[o11], w11, v);
        // output is streaming & never re-read: NT store keeps L2 for the gather
        __builtin_nontemporal_store(v, ob + c * HWn);
    }
}

extern "C" void kernel_launch(void* const* d_in, const int* in_sizes, int n_in,
                              void* d_out, int out_size, void* d_ws, size_t ws_size,
                              hipStream_t stream) {
    const float* x = (const float*)d_in[0];   // (4,64,512,512) f32
    const float* H = (const float*)d_in[1];   // (4,3,3) f32
    float* out = (float*)d_out;               // (4,64,512,512) f32

    dim3 grid(HWn / 256, Bn);                 // 1024 x 4 blocks
    dim3 block(256);                          // 8 wave32 waves per block
    homography_warp_kernel<<<grid, block, 0, stream>>>(x, H, out);
}